// Model_77592879170010
// MI455X (gfx1250) — compile-verified
//
#include <hip/hip_runtime.h>
#include <math.h>

// ---------------------------------------------------------------------------
// Transformer-XL forward on gfx1250 (MI455X), fp32 end-to-end.
// All GEMMs run on V_WMMA_F32_16X16X4_F32 (wave32 matrix engine, f32 in/out).
// LDS layouts chosen so every WMMA fragment is a single aligned ds_load_b64.
// ---------------------------------------------------------------------------

typedef float v2f __attribute__((ext_vector_type(2)));
typedef float v8f __attribute__((ext_vector_type(8)));

#define QLEN   2048
#define MLEN   1024
#define KLEN_  3072
#define DMODEL 1024
#define NHEAD  5
#define DHEAD  128
#define DINNER 4096

// ---------------------------------------------------------------------------
// Generic WMMA GEMM:  C[M,N] = (A[M,K] (+aBias[k])) @ B  (+cBias[n], relu?)
//   - 256 threads = 8 waves; wave grid 4(M) x 2(N); each wave 32x64 output
//   - block tile 128x128, K staged 32 at a time through LDS
//   - transB==1: B source is [N,K] row-major with leading dim ldb
// Requires: M%128==0, N%128==0, K%32==0 (true for every GEMM in this model)
// ---------------------------------------------------------------------------
__global__ __launch_bounds__(256)
void gemm_wmma_f32(const float* __restrict__ A, int lda,
                   const float* __restrict__ B, int ldb, int transB,
                   float* __restrict__ C, int ldc, int K,
                   const float* __restrict__ aBias,
                   const float* __restrict__ cBias, int relu)
{
  // A tile: 128 rows x 32 K; pitch 34 (even) so [m][even k] is 8B-aligned.
  __shared__ float As[128][34];
  // B tile: K-pairs interleaved: Bs2[kp][n] = {B[2kp][n], B[2kp+1][n]}.
  __shared__ v2f Bs2[16][129];

  const int tid  = threadIdx.x;
  const int lane = tid & 31;
  const int wave = tid >> 5;
  const int wm   = wave >> 1;     // 0..3  (M direction)
  const int wn   = wave & 1;      // 0..1  (N direction)
  const int lm   = lane & 15;
  const int lh   = lane >> 4;     // lane half
  const int row0 = blockIdx.y * 128;
  const int col0 = blockIdx.x * 128;

  v8f acc[2][4];
  #pragma unroll
  for (int i = 0; i < 2; ++i)
    #pragma unroll
    for (int j = 0; j < 4; ++j)
      #pragma unroll
      for (int r = 0; r < 8; ++r)
        acc[i][j][r] = 0.0f;

  for (int k0 = 0; k0 < K; k0 += 32) {
    // ---- prefetch next K-stage into cache (global_prefetch_b8) ----
    if (k0 + 32 < K) {
      if (tid < 128)
        __builtin_prefetch(A + (size_t)(row0 + tid) * lda + k0 + 32, 0, 1);
      if (!transB) {
        if (tid < 32)
          __builtin_prefetch(B + (size_t)(k0 + 32 + tid) * ldb + col0, 0, 1);
      } else {
        if (tid < 128)
          __builtin_prefetch(B + (size_t)(col0 + tid) * ldb + k0 + 32, 0, 1);
      }
    }

    // ---- stage A tile: 128 x 32 (float4 loads; optional per-K bias) ----
    #pragma unroll
    for (int it = 0; it < 4; ++it) {
      int idx = tid + it * 256;               // 0..1023
      int r   = idx >> 3;
      int c4  = (idx & 7) << 2;
      float4 a4 = *(const float4*)(A + (size_t)(row0 + r) * lda + k0 + c4);
      if (aBias) {
        a4.x += aBias[k0 + c4 + 0];
        a4.y += aBias[k0 + c4 + 1];
        a4.z += aBias[k0 + c4 + 2];
        a4.w += aBias[k0 + c4 + 3];
      }
      *(v2f*)&As[r][c4 + 0] = v2f{a4.x, a4.y};
      *(v2f*)&As[r][c4 + 2] = v2f{a4.z, a4.w};
    }
    // ---- stage B tile: 32 x 128 (pair-interleaved) ----
    if (!transB) {
      #pragma unroll
      for (int it = 0; it < 4; ++it) {
        int idx = tid + it * 256;
        int r   = idx >> 5;                   // 0..31 (K)
        int c4  = (idx & 31) << 2;            // 0..124 (N)
        float4 b4 = *(const float4*)(B + (size_t)(k0 + r) * ldb + col0 + c4);
        int kp = r >> 1, ke = r & 1;
        Bs2[kp][c4 + 0][ke] = b4.x;
        Bs2[kp][c4 + 1][ke] = b4.y;
        Bs2[kp][c4 + 2][ke] = b4.z;
        Bs2[kp][c4 + 3][ke] = b4.w;
      }
    } else {
      #pragma unroll
      for (int it = 0; it < 4; ++it) {
        int idx = tid + it * 256;
        int n   = idx >> 3;                   // 0..127 (N rows of source)
        int c4  = (idx & 7) << 2;             // 0..28  (K)
        float4 b4 = *(const float4*)(B + (size_t)(col0 + n) * ldb + k0 + c4);
        Bs2[(c4 >> 1) + 0][n] = v2f{b4.x, b4.y};
        Bs2[(c4 >> 1) + 1][n] = v2f{b4.z, b4.w};
      }
    }
    __syncthreads();

    // ---- 8 K-steps of 4, 2x4 WMMA tiles per wave ----
    #pragma unroll
    for (int ks = 0; ks < 8; ++ks) {
      const int kb = (ks << 2) + (lh << 1);   // K base for this lane half (even)
      v2f a[2], b[4];
      #pragma unroll
      for (int tm = 0; tm < 2; ++tm) {
        int m = (wm << 5) + (tm << 4) + lm;
        a[tm] = *(const v2f*)&As[m][kb];      // one ds_load_b64
      }
      #pragma unroll
      for (int tn = 0; tn < 4; ++tn) {
        int n = (wn << 6) + (tn << 4) + lm;
        b[tn] = Bs2[kb >> 1][n];              // one ds_load_b64
      }
      #pragma unroll
      for (int tm = 0; tm < 2; ++tm)
        #pragma unroll
        for (int tn = 0; tn < 4; ++tn)
          acc[tm][tn] = __builtin_amdgcn_wmma_f32_16x16x4_f32(
              false, a[tm], false, b[tn], (short)0, acc[tm][tn], false, false);
    }
    __syncthreads();
  }

  // ---- epilogue: C/D layout M = r + 8*lh, N = lm ----
  #pragma unroll
  for (int tm = 0; tm < 2; ++tm) {
    #pragma unroll
    for (int tn = 0; tn < 4; ++tn) {
      int mb = row0 + (wm << 5) + (tm << 4) + (lh << 3);
      int n  = col0 + (wn << 6) + (tn << 4) + lm;
      float cb = cBias ? cBias[n] : 0.0f;
      #pragma unroll
      for (int r = 0; r < 8; ++r) {
        float v = acc[tm][tn][r] + cb;
        if (relu) v = fmaxf(v, 0.0f);
        C[(size_t)(mb + r) * ldc + n] = v;
      }
    }
  }
}

// ---------------------------------------------------------------------------
// Row-wise helper kernels (wave32 shuffles + small LDS reductions)
// ---------------------------------------------------------------------------
__device__ inline float block_sum_256(float v, float* sred) {
  #pragma unroll
  for (int off = 16; off > 0; off >>= 1) v += __shfl_xor(v, off, 32);
  if ((threadIdx.x & 31) == 0) sred[threadIdx.x >> 5] = v;
  __syncthreads();
  if (threadIdx.x == 0) {
    float s = 0.0f;
    for (int i = 0; i < 8; ++i) s += sred[i];
    sred[0] = s;
  }
  __syncthreads();
  float r = sred[0];
  __syncthreads();
  return r;
}

__device__ inline float block_max_256(float v, float* sred) {
  #pragma unroll
  for (int off = 16; off > 0; off >>= 1) v = fmaxf(v, __shfl_xor(v, off, 32));
  if ((threadIdx.x & 31) == 0) sred[threadIdx.x >> 5] = v;
  __syncthreads();
  if (threadIdx.x == 0) {
    float s = sred[0];
    for (int i = 1; i < 8; ++i) s = fmaxf(s, sred[i]);
    sred[0] = s;
  }
  __syncthreads();
  float r = sred[0];
  __syncthreads();
  return r;
}

// h[q,:] = word_emb[ids[q],:] * sqrt(DMODEL)
__global__ __launch_bounds__(256)
void embed_k(const int* __restrict__ ids, const float* __restrict__ emb,
             float* __restrict__ H)
{
  int q  = blockIdx.x;
  int id = ids[q];
  const float* src = emb + (size_t)id * DMODEL;
  float* dst = H + (size_t)q * DMODEL;
  for (int d = threadIdx.x; d < DMODEL; d += 256)
    dst[d] = src[d] * 32.0f;   // sqrt(1024)
}

// pos_emb[k,:] : pos = min(KLEN-1-k, 1000); [sin(pos*w) | cos(pos*w)]
__global__ __launch_bounds__(256)
void posemb_k(float* __restrict__ P)
{
  int k = blockIdx.x;
  float pos = fminf((float)(KLEN_ - 1 - k), 1000.0f);
  float* dst = P + (size_t)k * DMODEL;
  for (int d = threadIdx.x; d < DMODEL; d += 256) {
    int   j    = d & 511;
    float invf = expf(-(float)j * (9.210340371976184f / 512.0f)); // 10000^(-j/512)
    float v    = pos * invf;
    dst[d] = (d < 512) ? sinf(v) : cosf(v);
  }
}

__global__ void copy_k(const float* __restrict__ src, float* __restrict__ dst, int n)
{
  int i = blockIdx.x * blockDim.x + threadIdx.x;
  if (i < n) dst[i] = src[i];
}

// S[q,j] = (j <= q+MLEN) ? (AC[q,j] + BD[q, j + QLEN-1 - q]) / sqrt(DH) : -1e30
// (closed-form TransfoXL rel_shift on the unmasked region)
__global__ __launch_bounds__(256)
void combine_k(float* __restrict__ S, const float* __restrict__ BD)
{
  int q = blockIdx.x;
  const float scale = 0.08838834764831845f;  // 1/sqrt(128)
  float* row = S + (size_t)q * KLEN_;
  const float* bd = BD + (size_t)q * KLEN_;
  for (int j = threadIdx.x; j < KLEN_; j += 256) {
    float v;
    if (j <= q + MLEN) v = (row[j] + bd[j + (QLEN - 1) - q]) * scale;
    else               v = -1e30f;
    row[j] = v;
  }
}

__global__ __launch_bounds__(256)
void softmax_k(float* __restrict__ S)
{
  __shared__ float sred[8];
  int q = blockIdx.x;
  float* row = S + (size_t)q * KLEN_;
  float mx = -3.402823e38f;
  for (int j = threadIdx.x; j < KLEN_; j += 256) mx = fmaxf(mx, row[j]);
  mx = block_max_256(mx, sred);
  float sum = 0.0f;
  for (int j = threadIdx.x; j < KLEN_; j += 256) {
    float e = __expf(row[j] - mx);
    row[j] = e;
    sum += e;
  }
  sum = block_sum_256(sum, sred);
  float inv = 1.0f / sum;
  for (int j = threadIdx.x; j < KLEN_; j += 256) row[j] *= inv;
}

// H[q,:] = LayerNorm(H[q,:] + X[q,:]) * g + b    (in place on H)
__global__ __launch_bounds__(256)
void resln_k(float* __restrict__ H, const float* __restrict__ X,
             const float* __restrict__ g, const float* __restrict__ b)
{
  __shared__ float sred[8];
  int q = blockIdx.x;
  float v[4];
  #pragma unroll
  for (int t = 0; t < 4; ++t) {
    int d = threadIdx.x + (t << 8);
    v[t] = H[(size_t)q * DMODEL + d] + X[(size_t)q * DMODEL + d];
  }
  float mean = block_sum_256(v[0] + v[1] + v[2] + v[3], sred) * (1.0f / DMODEL);
  float sq = 0.0f;
  #pragma unroll
  for (int t = 0; t < 4; ++t) { v[t] -= mean; sq += v[t] * v[t]; }
  float var = block_sum_256(sq, sred) * (1.0f / DMODEL);
  float inv = rsqrtf(var + 1e-5f);
  #pragma unroll
  for (int t = 0; t < 4; ++t) {
    int d = threadIdx.x + (t << 8);
    H[(size_t)q * DMODEL + d] = v[t] * inv * g[d] + b[d];
  }
}

// preds[q,:] = H[q,:] @ out_w[1024,2] + out_b
__global__ __launch_bounds__(256)
void preds_k(const float* __restrict__ H, const float* __restrict__ W,
             const float* __restrict__ bias, float* __restrict__ out)
{
  __shared__ float sred[8];
  int q = blockIdx.x;
  float s0 = 0.0f, s1 = 0.0f;
  for (int k = threadIdx.x; k < DMODEL; k += 256) {
    float h = H[(size_t)q * DMODEL + k];
    s0 += h * W[k * 2 + 0];
    s1 += h * W[k * 2 + 1];
  }
  s0 = block_sum_256(s0, sred);
  s1 = block_sum_256(s1, sred);
  if (threadIdx.x == 0) {
    out[q * 2 + 0] = s0 + bias[0];
    out[q * 2 + 1] = s1 + bias[1];
  }
}

// ---------------------------------------------------------------------------
// Host orchestration
// ---------------------------------------------------------------------------
static inline void launch_gemm(hipStream_t s, const float* A, int lda,
                               const float* B, int ldb, int transB,
                               float* C, int ldc, int M, int N, int K,
                               const float* aBias, const float* cBias, int relu)
{
  dim3 grid(N / 128, M / 128);
  gemm_wmma_f32<<<grid, 256, 0, s>>>(A, lda, B, ldb, transB, C, ldc, K,
                                     aBias, cBias, relu);
}

extern "C" void kernel_launch(void* const* d_in, const int* in_sizes, int n_in,
                              void* d_out, int out_size, void* d_ws, size_t ws_size,
                              hipStream_t stream)
{
  (void)in_sizes; (void)n_in; (void)out_size; (void)ws_size;

  const int*   ids      = (const int*)d_in[0];
  const float* mems     = (const float*)d_in[1];   // [3,1024,1,1024]
  const float* word_emb = (const float*)d_in[2];   // [VOCAB,1024]
  const float* qkv_w    = (const float*)d_in[3];   // [3,1024,1920]
  const float* o_w      = (const float*)d_in[4];   // [3,640,1024]
  const float* r_w      = (const float*)d_in[5];   // [3,1024,640]
  const float* r_w_bias = (const float*)d_in[6];   // [3,5,128]
  const float* r_r_bias = (const float*)d_in[7];   // [3,5,128]
  const float* ln1_g    = (const float*)d_in[8];
  const float* ln1_b    = (const float*)d_in[9];
  const float* ff_w1    = (const float*)d_in[10];  // [3,1024,4096]
  const float* ff_b1    = (const float*)d_in[11];  // [3,4096]
  const float* ff_w2    = (const float*)d_in[12];  // [3,4096,1024]
  const float* ff_b2    = (const float*)d_in[13];  // [3,1024]
  const float* ln2_g    = (const float*)d_in[14];
  const float* ln2_b    = (const float*)d_in[15];
  const float* out_w    = (const float*)d_in[16];  // [1024,2]
  const float* out_b    = (const float*)d_in[17];  // [2]

  float* out       = (float*)d_out;
  float* out_preds = out;                                  // 2048*2
  float* out_mems  = out + (size_t)QLEN * 2;               // 3*1024*1024
  float* out_att0  = out_mems + (size_t)3 * MLEN * DMODEL; // 2048*3072

  // workspace layout (floats); total ~34.4M floats (~137 MB)
  float* ws = (float*)d_ws;
  size_t o = 0;
  float* H     = ws + o; o += (size_t)QLEN * DMODEL;    // hidden [2048,1024]
  float* CAT   = ws + o; o += (size_t)KLEN_ * DMODEL;   // [3072,1024]
  float* HEADS = ws + o; o += (size_t)KLEN_ * 3 * NHEAD * DHEAD; // [3072,1920]
  float* RH    = ws + o; o += (size_t)KLEN_ * NHEAD * DHEAD;     // [3072,640]
  float* SCORE = ws + o; o += (size_t)QLEN * KLEN_;     // [2048,3072]
  float* AV    = ws + o; o += (size_t)QLEN * NHEAD * DHEAD;      // [2048,640]
  float* TMP   = ws + o; o += (size_t)QLEN * DMODEL;    // [2048,1024]
  float* FF1   = ws + o; o += (size_t)QLEN * DINNER;    // [2048,4096]; doubles as BD
  float* POS   = ws + o; o += (size_t)KLEN_ * DMODEL;   // [3072,1024]
  float* BD    = FF1;  // attention phase and FF phase never overlap

  auto copy = [&](const float* src, float* dst, int n) {
    copy_k<<<(n + 255) / 256, 256, 0, stream>>>(src, dst, n);
  };

  embed_k<<<QLEN, 256, 0, stream>>>(ids, word_emb, H);
  posemb_k<<<KLEN_, 256, 0, stream>>>(POS);

  for (int i = 0; i < 3; ++i) {
    // new_mems[i] = concat(mems[i], h)[-MLEN:] == h rows [QLEN-MLEN, QLEN)
    copy(H + (size_t)(QLEN - MLEN) * DMODEL,
         out_mems + (size_t)i * MLEN * DMODEL, MLEN * DMODEL);

    // cat = [mems[i]; h]
    copy(mems + (size_t)i * MLEN * DMODEL, CAT, MLEN * DMODEL);
    copy(H, CAT + (size_t)MLEN * DMODEL, QLEN * DMODEL);

    // heads = cat @ qkv_w[i]  -> [3072,1920]  (q | k | v, each [*,5,128])
    launch_gemm(stream, CAT, DMODEL,
                qkv_w + (size_t)i * DMODEL * 3 * NHEAD * DHEAD, 3 * NHEAD * DHEAD, 0,
                HEADS, 3 * NHEAD * DHEAD, KLEN_, 3 * NHEAD * DHEAD, DMODEL,
                nullptr, nullptr, 0);

    // rh = pos_emb @ r_w[i] -> [3072,640]
    launch_gemm(stream, POS, DMODEL,
                r_w + (size_t)i * DMODEL * NHEAD * DHEAD, NHEAD * DHEAD, 0,
                RH, NHEAD * DHEAD, KLEN_, NHEAD * DHEAD, DMODEL,
                nullptr, nullptr, 0);

    const float* Q = HEADS + (size_t)MLEN * 3 * NHEAD * DHEAD;          // q rows
    for (int n = 0; n < NHEAD; ++n) {
      const float* Qh = Q + n * DHEAD;                                  // cols [n*128)
      const float* Kh = HEADS + NHEAD * DHEAD + n * DHEAD;              // k section
      const float* Vh = HEADS + 2 * NHEAD * DHEAD + n * DHEAD;          // v section

      // AC = (q + r_w_bias) @ k^T  -> SCORE [2048,3072]
      launch_gemm(stream, Qh, 3 * NHEAD * DHEAD,
                  Kh, 3 * NHEAD * DHEAD, 1,
                  SCORE, KLEN_, QLEN, KLEN_, DHEAD,
                  r_w_bias + (size_t)(i * NHEAD + n) * DHEAD, nullptr, 0);

      // BD = (q + r_r_bias) @ rh^T -> BD [2048,3072]
      launch_gemm(stream, Qh, 3 * NHEAD * DHEAD,
                  RH + n * DHEAD, NHEAD * DHEAD, 1,
                  BD, KLEN_, QLEN, KLEN_, DHEAD,
                  r_r_bias + (size_t)(i * NHEAD + n) * DHEAD, nullptr, 0);

      // score = mask((AC + rel_shift(BD)) * scale); softmax over k
      combine_k<<<QLEN, 256, 0, stream>>>(SCORE, BD);
      softmax_k<<<QLEN, 256, 0, stream>>>(SCORE);

      if (i == 0 && n == 0)
        copy(SCORE, out_att0, QLEN * KLEN_);

      // av[:, n*128:(n+1)*128] = prob @ v
      launch_gemm(stream, SCORE, KLEN_,
                  Vh, 3 * NHEAD * DHEAD, 0,
                  AV + n * DHEAD, NHEAD * DHEAD, QLEN, DHEAD, KLEN_,
                  nullptr, nullptr, 0);
    }

    // TMP = av @ o_w[i];  H = LN(H + TMP)
    launch_gemm(stream, AV, NHEAD * DHEAD,
                o_w + (size_t)i * NHEAD * DHEAD * DMODEL, DMODEL, 0,
                TMP, DMODEL, QLEN, DMODEL, NHEAD * DHEAD,
                nullptr, nullptr, 0);
    resln_k<<<QLEN, 256, 0, stream>>>(H, TMP, ln1_g + i * DMODEL, ln1_b + i * DMODEL);

    // FF1 = relu(H @ ff_w1 + b1);  TMP = FF1 @ ff_w2 + b2;  H = LN(H + TMP)
    launch_gemm(stream, H, DMODEL,
                ff_w1 + (size_t)i * DMODEL * DINNER, DINNER, 0,
                FF1, DINNER, QLEN, DINNER, DMODEL,
                nullptr, ff_b1 + (size_t)i * DINNER, 1);
    launch_gemm(stream, FF1, DINNER,
                ff_w2 + (size_t)i * DINNER * DMODEL, DMODEL, 0,
                TMP, DMODEL, QLEN, DMODEL, DINNER,
                nullptr, ff_b2 + (size_t)i * DMODEL, 0);
    resln_k<<<QLEN, 256, 0, stream>>>(H, TMP, ln2_g + i * DMODEL, ln2_b + i * DMODEL);
  }

  preds_k<<<QLEN, 256, 0, stream>>>(H, out_w, out_b, out_preds);
}